// GlobalAggregation_7206955122837
// MI455X (gfx1250) — compile-verified
//
#include <hip/hip_runtime.h>
#include <hip/hip_bf16.h>
#include <math.h>

// ---------------------------------------------------------------------------
// GlobalAggregation: segment {sum,mean,max} pool (sorted batch) + 2-layer MLP
//   x:[N,128] f32, batch:[N] i32 sorted, G=4096 graphs
//   H = concat(max, mean, sum) [G,384]
//   h = leaky_relu(H @ W1^T + b1) ; out = h @ W2^T + b2   -> [G,128] f32
// Memory-bound on streaming x (~1.02 GB); MLP done with v_wmma_f32_16x16x4_f32.
// ---------------------------------------------------------------------------

#define DEPTH 128
#define NPB   256   // nodes per pooling block

typedef __attribute__((ext_vector_type(2))) float v2f;
typedef __attribute__((ext_vector_type(8))) float v8f;

// Exact float atomic-max via integer atomics (valid with -inf initialization).
__device__ __forceinline__ void atomicMaxFloat(float* addr, float val) {
    if (val >= 0.0f) {
        atomicMax((int*)addr, __float_as_int(val));
    } else {
        atomicMin((unsigned int*)addr, __float_as_uint(val));
    }
}

__global__ void ga_init_kernel(float* __restrict__ sum, float* __restrict__ mx,
                               float* __restrict__ cnt, int G) {
    int i = blockIdx.x * blockDim.x + threadIdx.x;
    int total = G * DEPTH;
    if (i < total) {
        sum[i] = 0.0f;
        mx[i]  = -INFINITY;
    }
    if (i < G) cnt[i] = 0.0f;
}

// One thread per channel; each block walks NPB consecutive (sorted) rows and
// flushes running sum/max to global atomics only at segment boundaries.
__global__ __launch_bounds__(DEPTH) void ga_pool_kernel(
        const float* __restrict__ x, const int* __restrict__ batch,
        float* __restrict__ sum, float* __restrict__ mx,
        float* __restrict__ cnt, int nNodes) {
    const int t = threadIdx.x;                 // channel 0..127
    long long start = (long long)blockIdx.x * NPB;
    long long end   = start + NPB;
    if (end > nNodes) end = nNodes;
    if (start >= end) return;

    int   curg = batch[start];
    float rs   = 0.0f;
    float rm   = -INFINITY;
    int   run  = 0;

    for (long long i = start; i < end; ++i) {
        int g = batch[i];
        if (g != curg) {
            atomicAdd(&sum[(size_t)curg * DEPTH + t], rs);
            atomicMaxFloat(&mx[(size_t)curg * DEPTH + t], rm);
            if (t == 0) atomicAdd(&cnt[curg], (float)run);
            rs = 0.0f; rm = -INFINITY; run = 0; curg = g;
        }
        float v = x[i * DEPTH + t];
        rs += v;
        rm = fmaxf(rm, v);
        ++run;
    }
    atomicAdd(&sum[(size_t)curg * DEPTH + t], rs);
    atomicMaxFloat(&mx[(size_t)curg * DEPTH + t], rm);
    if (t == 0) atomicAdd(&cnt[curg], (float)run);
}

// H[g, 0:128)=max (0 if empty), [128:256)=mean, [256:384)=sum
__global__ void ga_finalize_kernel(const float* __restrict__ sum,
                                   const float* __restrict__ mx,
                                   const float* __restrict__ cnt,
                                   float* __restrict__ H, int G) {
    int idx = blockIdx.x * blockDim.x + threadIdx.x;
    int total = G * 3 * DEPTH;
    if (idx >= total) return;
    int g = idx / (3 * DEPTH);
    int k = idx % (3 * DEPTH);
    float c = cnt[g];
    float v;
    if (k < DEPTH) {
        v = (c > 0.0f) ? mx[(size_t)g * DEPTH + k] : 0.0f;
    } else if (k < 2 * DEPTH) {
        v = sum[(size_t)g * DEPTH + (k - DEPTH)] / fmaxf(c, 1.0f);
    } else {
        v = sum[(size_t)g * DEPTH + (k - 2 * DEPTH)];
    }
    H[idx] = v;
}

// Fused MLP: 8 waves/block; wave w owns output-channel tile [16w, 16w+16).
// Layer 1: [16 x 384] @ [384 x 16] via v_wmma_f32_16x16x4_f32 (K-loop step 4),
// bias + LeakyReLU -> LDS band [16 x 128]; barrier; Layer 2 from LDS.
__global__ __launch_bounds__(256) void ga_mlp_kernel(
        const float* __restrict__ H,
        const float* __restrict__ W1, const float* __restrict__ b1,
        const float* __restrict__ W2, const float* __restrict__ b2,
        float* __restrict__ out, int G) {
    __shared__ float hsh[16 * DEPTH];   // 8 KB

    const int tid  = threadIdx.x;
    const int wave = tid >> 5;          // 0..7  -> N tile
    const int lane = tid & 31;
    const int m0   = blockIdx.x * 16;   // graph-row base
    const int r    = lane & 15;         // row (A) / col (B,C,D) within tile
    const int kb   = (lane >> 4) << 1;  // 0 for lanes 0-15, 2 for lanes 16-31
    const int rowh = ((lane >> 4) << 3);// +8 row offset for upper half (C/D)
    const int n0   = wave * 16;

    // ---- layer 1: acc = H[m0:m0+16, :384] x W1^T[:, n0:n0+16] ----
    v8f acc = {};
    const float* Hrow  = H  + (size_t)(m0 + r) * (3 * DEPTH);
    const float* W1row = W1 + (size_t)(n0 + r) * (3 * DEPTH);  // B col = W1 row
    for (int k = 0; k < 3 * DEPTH; k += 4) {
        v2f a; a.x = Hrow[k + kb];  a.y = Hrow[k + kb + 1];
        v2f b; b.x = W1row[k + kb]; b.y = W1row[k + kb + 1];
        acc = __builtin_amdgcn_wmma_f32_16x16x4_f32(
            /*neg_a=*/false, a, /*neg_b=*/false, b,
            /*c_mod=*/(short)0, acc, /*reuse_a=*/false, /*reuse_b=*/false);
    }

    const int colOut = n0 + r;
    const float bb1 = b1[colOut];
#pragma unroll
    for (int j = 0; j < 8; ++j) {
        int row = j + rowh;             // 0..7 or 8..15
        float v = acc[j] + bb1;
        v = (v > 0.0f) ? v : 0.01f * v; // LeakyReLU slope 0.01
        hsh[row * DEPTH + colOut] = v;
    }
    __syncthreads();

    // ---- layer 2: acc2 = h[m0 band, :128] x W2^T[:, n0:n0+16] ----
    v8f acc2 = {};
    const float* W2row = W2 + (size_t)(n0 + r) * DEPTH;
    for (int k = 0; k < DEPTH; k += 4) {
        v2f a; a.x = hsh[r * DEPTH + k + kb]; a.y = hsh[r * DEPTH + k + kb + 1];
        v2f b; b.x = W2row[k + kb];           b.y = W2row[k + kb + 1];
        acc2 = __builtin_amdgcn_wmma_f32_16x16x4_f32(
            false, a, false, b, (short)0, acc2, false, false);
    }

    const float bb2 = b2[colOut];
#pragma unroll
    for (int j = 0; j < 8; ++j) {
        int row = j + rowh;
        out[(size_t)(m0 + row) * DEPTH + colOut] = acc2[j] + bb2;
    }
    (void)G;
}

extern "C" void kernel_launch(void* const* d_in, const int* in_sizes, int n_in,
                              void* d_out, int out_size, void* d_ws, size_t ws_size,
                              hipStream_t stream) {
    const float* x     = (const float*)d_in[0];
    const int*   batch = (const int*)  d_in[1];
    // d_in[2] = output_dim scalar (device); G derived from out_size instead.
    const float* W1    = (const float*)d_in[3];
    const float* b1    = (const float*)d_in[4];
    const float* W2    = (const float*)d_in[5];
    const float* b2    = (const float*)d_in[6];
    float*       out   = (float*)d_out;

    const int nNodes = in_sizes[1];        // 2,000,000
    const int G      = out_size / DEPTH;   // 4096

    // Workspace layout (floats): sum[G*128] | mx[G*128] | cnt[G] | H[G*384]
    float* ws  = (float*)d_ws;
    float* sum = ws;
    float* mx  = sum + (size_t)G * DEPTH;
    float* cnt = mx  + (size_t)G * DEPTH;
    float* H   = cnt + (size_t)G;

    {
        int total = G * DEPTH;
        ga_init_kernel<<<(total + 255) / 256, 256, 0, stream>>>(sum, mx, cnt, G);
    }
    {
        int blocks = (nNodes + NPB - 1) / NPB;
        ga_pool_kernel<<<blocks, DEPTH, 0, stream>>>(x, batch, sum, mx, cnt, nNodes);
    }
    {
        int total = G * 3 * DEPTH;
        ga_finalize_kernel<<<(total + 255) / 256, 256, 0, stream>>>(sum, mx, cnt, H, G);
    }
    {
        ga_mlp_kernel<<<G / 16, 256, 0, stream>>>(H, W1, b1, W2, b2, out, G);
    }
}